// CF_42116449305198
// MI455X (gfx1250) — compile-verified
//
#include <hip/hip_runtime.h>
#include <math.h>

#define N_USERS 8192
#define N_ITEMS 4096
#define BATCH   8192
#define G_SPLIT 8
#define K_PER_G (N_USERS / G_SPLIT)   // 1024 users of the K-dim per wave

typedef __attribute__((ext_vector_type(2))) float v2f;
typedef __attribute__((ext_vector_type(8))) float v8f;

// ---------------------------------------------------------------------------
// K1: per-user mean over nonzero entries of rating row.  One wave per row.
// Streams R once (128 MB).  ub[u] = cnt>0 ? row_sum/cnt : 0
// ---------------------------------------------------------------------------
__global__ void user_mean_kernel(const float* __restrict__ R,
                                 float* __restrict__ ub) {
    int wave = (blockIdx.x * blockDim.x + threadIdx.x) >> 5;
    int lane = threadIdx.x & 31;
    if (wave >= N_USERS) return;
    const float4* row = reinterpret_cast<const float4*>(R + (size_t)wave * N_ITEMS);
    float sum = 0.f;
    int   cnt = 0;
    for (int j = lane; j < N_ITEMS / 4; j += 32) {
        float4 v = row[j];
        sum += v.x + v.y + v.z + v.w;
        cnt += (v.x != 0.f) + (v.y != 0.f) + (v.z != 0.f) + (v.w != 0.f);
    }
    // wave32 shuffle reduction
    for (int off = 16; off > 0; off >>= 1) {
        sum += __shfl_down(sum, off, 32);
        cnt += __shfl_down(cnt, off, 32);
    }
    if (lane == 0) ub[wave] = (cnt > 0) ? sum / (float)cnt : 0.f;
}

// ---------------------------------------------------------------------------
// K2: AT[i][u] = R[u][i] - ub[u]   (LDS-tiled 32x32 transpose, coalesced both ways)
// ---------------------------------------------------------------------------
__global__ void transpose_adjust_kernel(const float* __restrict__ R,
                                        const float* __restrict__ ub,
                                        float* __restrict__ AT) {
    __shared__ float tile[32][33];          // +1 pad: conflict-free transpose
    int tx = threadIdx.x, ty = threadIdx.y;
    int i0 = blockIdx.x * 32;               // item tile
    int u0 = blockIdx.y * 32;               // user tile
#pragma unroll
    for (int rr = ty; rr < 32; rr += 8) {
        int u = u0 + rr;
        tile[rr][tx] = R[(size_t)u * N_ITEMS + i0 + tx] - ub[u];
    }
    __syncthreads();
#pragma unroll
    for (int rr = ty; rr < 32; rr += 8) {
        AT[(size_t)(i0 + rr) * N_USERS + u0 + tx] = tile[tx][rr];
    }
}

// ---------------------------------------------------------------------------
// K3: batched dots via V_WMMA_F32_16X16X4_F32.
// One wave (block of 32) handles 16 batch elements over a 1024-long K slice.
//   A[m][k] = S[user[bbase+m]][k]         (16x4 per step)
//   B[k][n] = AT[item[bbase+n]][k]        (4x16 per step)
// ISA VGPR layout for both A(16x4) and B(4x16) f32:
//   lanes 0-15 hold K={k,k+1} in v0,v1 ; lanes 16-31 hold K={k+2,k+3}
// -> identical per-lane float2 gathered loads for A and B.
// Diagonal D[m][m] of the chained accumulator is the 16 dot products.
// ---------------------------------------------------------------------------
template <bool USE_AT>
__global__ void score_kernel(const int* __restrict__ user,
                             const int* __restrict__ item,
                             const float* __restrict__ S,
                             const float* __restrict__ Bsrc,   // AT (transposed) or R (fallback)
                             const float* __restrict__ ub,
                             float* __restrict__ partial) {
    int wid   = blockIdx.x;                  // blockDim.x == 32 : one wave per block
    int grp   = wid & (G_SPLIT - 1);
    int bbase = (wid / G_SPLIT) * 16;
    int l     = threadIdx.x;
    int r     = l & 15;                      // matrix row/col handled by this lane
    int koff  = (l >> 4) * 2;                // lanes 16-31 carry K+2,K+3

    int u_idx = user[bbase + r];
    int i_idx = item[bbase + r];
    const float* arow = S + (size_t)u_idx * N_USERS;
    const float* brow = USE_AT ? (Bsrc + (size_t)i_idx * N_USERS) : nullptr;

    int k0 = grp * K_PER_G;
    v8f c0 = {};
    v8f c1 = {};
    for (int k = k0; k < k0 + K_PER_G; k += 8) {
        int ka = k + koff;
        int kb = k + 4 + koff;
        v2f a0 = *reinterpret_cast<const v2f*>(arow + ka);
        v2f a1 = *reinterpret_cast<const v2f*>(arow + kb);
        v2f b0, b1;
        if (USE_AT) {
            b0 = *reinterpret_cast<const v2f*>(brow + ka);
            b1 = *reinterpret_cast<const v2f*>(brow + kb);
        } else {
            // fallback: strided column walk of R, adjust on the fly
            b0[0] = Bsrc[(size_t)(ka    ) * N_ITEMS + i_idx] - ub[ka];
            b0[1] = Bsrc[(size_t)(ka + 1) * N_ITEMS + i_idx] - ub[ka + 1];
            b1[0] = Bsrc[(size_t)(kb    ) * N_ITEMS + i_idx] - ub[kb];
            b1[1] = Bsrc[(size_t)(kb + 1) * N_ITEMS + i_idx] - ub[kb + 1];
        }
        // two independent accumulator chains hide WMMA latency
        c0 = __builtin_amdgcn_wmma_f32_16x16x4_f32(false, a0, false, b0,
                                                   (short)0, c0, false, false);
        c1 = __builtin_amdgcn_wmma_f32_16x16x4_f32(false, a1, false, b1,
                                                   (short)0, c1, false, false);
    }
    v8f c = c0 + c1;

    // Diagonal extraction.  C/D layout: VGPR v, lanes 0-15 -> (M=v, N=lane),
    // lanes 16-31 -> (M=v+8, N=lane-16).  So diag m lives at:
    //   m in 0..7  : c[m]    in lane m
    //   m in 8..15 : c[m-8]  in lane m+16
    int ci = (l < 16) ? l : (l - 24);        // meaningful only for l<8 or l>=24
    float v = c[0];
    v = (ci == 1) ? c[1] : v;
    v = (ci == 2) ? c[2] : v;
    v = (ci == 3) ? c[3] : v;
    v = (ci == 4) ? c[4] : v;
    v = (ci == 5) ? c[5] : v;
    v = (ci == 6) ? c[6] : v;
    v = (ci == 7) ? c[7] : v;

    float* slot = partial + (size_t)grp * BATCH + bbase;
    if (l < 8)        slot[l]        = v;
    else if (l >= 24) slot[l - 16]   = v;
}

// ---------------------------------------------------------------------------
// K4: deterministic (fixed-order) combine of K-split partials + biases + sigmoid
// ---------------------------------------------------------------------------
__global__ void finalize_kernel(const int* __restrict__ user,
                                const int* __restrict__ item,
                                const float* __restrict__ partial,
                                const float* __restrict__ user_bias,
                                const float* __restrict__ item_bias,
                                const float* __restrict__ gb,
                                float* __restrict__ out) {
    int b = blockIdx.x * blockDim.x + threadIdx.x;
    if (b >= BATCH) return;
    float s = 0.f;
#pragma unroll
    for (int g = 0; g < G_SPLIT; ++g) s += partial[(size_t)g * BATCH + b];
    s += user_bias[user[b]] + item_bias[item[b]] + gb[0];
    out[b] = 5.0f / (1.0f + __expf(-s));
}

// ---------------------------------------------------------------------------
extern "C" void kernel_launch(void* const* d_in, const int* in_sizes, int n_in,
                              void* d_out, int out_size, void* d_ws, size_t ws_size,
                              hipStream_t stream) {
    const int*   user      = (const int*)d_in[0];
    const int*   item      = (const int*)d_in[1];
    const float* R         = (const float*)d_in[2];   // (8192, 4096)
    const float* S         = (const float*)d_in[3];   // (8192, 8192)
    const float* user_bias = (const float*)d_in[4];
    const float* item_bias = (const float*)d_in[5];
    const float* gb        = (const float*)d_in[6];
    float*       out       = (float*)d_out;

    // workspace layout: [ub: 32KB][partial: 256KB][pad to 512KB][AT: 128MB]
    float* ub      = (float*)d_ws;
    float* partial = ub + N_USERS;
    const size_t at_off = (size_t)512 * 1024;
    float* AT      = (float*)((char*)d_ws + at_off);
    const size_t need = at_off + (size_t)N_ITEMS * N_USERS * sizeof(float);
    const bool use_at = (ws_size >= need);

    user_mean_kernel<<<N_USERS / 8, 256, 0, stream>>>(R, ub);

    if (use_at) {
        dim3 blk(32, 8);
        dim3 grd(N_ITEMS / 32, N_USERS / 32);
        transpose_adjust_kernel<<<grd, blk, 0, stream>>>(R, ub, AT);
        score_kernel<true><<<(BATCH / 16) * G_SPLIT, 32, 0, stream>>>(
            user, item, S, AT, ub, partial);
    } else {
        score_kernel<false><<<(BATCH / 16) * G_SPLIT, 32, 0, stream>>>(
            user, item, S, R, ub, partial);
    }

    finalize_kernel<<<BATCH / 256, 256, 0, stream>>>(
        user, item, partial, user_bias, item_bias, gb, out);
}